// GATLayer_62680752718496
// MI455X (gfx1250) — compile-verified
//
#include <hip/hip_runtime.h>
#include <hip/hip_bf16.h>

typedef __attribute__((ext_vector_type(2))) float v2f;
typedef __attribute__((ext_vector_type(8))) float v8f;

#define IN_CH   256
#define HC      128   // HEADS*OUT_CH
#define HEADS   4
#define OUT_CH  32
#define LDS_STRIDE 260  // 256 + 4 pad -> bank-conflict-free column reads

// ---------- order-preserving float<->uint map for atomic max ----------
__device__ __forceinline__ unsigned mapf(float f) {
    unsigned u = __float_as_uint(f);
    return (u >> 31) ? ~u : (u | 0x80000000u);
}
__device__ __forceinline__ float unmapf(unsigned u) {
    return (u >> 31) ? __uint_as_float(u ^ 0x80000000u) : __uint_as_float(~u);
}
__device__ __forceinline__ float lrelu(float a) { return a > 0.0f ? a : 0.2f * a; }

// ---------- 0: init accumulators ----------
__global__ void gat_init(float* out_mean, unsigned* segmax, float* denom, int n) {
    int i = blockIdx.x * blockDim.x + threadIdx.x;
    if (i < n * OUT_CH) out_mean[i] = 0.0f;
    if (i < n * HEADS)  { segmax[i] = 0u; denom[i] = 0.0f; }
}

// ---------- 1: h = x @ W via V_WMMA_F32_16X16X4_F32 ----------
// grid.x = ceil(n/16); 256 threads = 8 waves; wave w -> N-tile [16w,16w+16)
__global__ void gat_gemm(const float* __restrict__ x, const float* __restrict__ W,
                         float* __restrict__ h, int n) {
    __shared__ float xs[16 * LDS_STRIDE];
    const int row0 = blockIdx.x * 16;

    // stage 16x256 fp32 tile of x into LDS with float4 loads
    const float4* xv = (const float4*)x;           // 64 float4 per row
    for (int i = 0; i < 4; ++i) {
        int idx = threadIdx.x + i * 256;           // 0..1023 float4 slots
        int r   = idx >> 6;                        // tile row
        int c4  = idx & 63;                        // float4 col
        int gr  = row0 + r;  if (gr >= n) gr = n - 1;
        float4 v = xv[(size_t)gr * 64 + c4];
        float* dst = &xs[r * LDS_STRIDE + (c4 << 2)];
        dst[0] = v.x; dst[1] = v.y; dst[2] = v.z; dst[3] = v.w;
    }
    __syncthreads();

    const int lane = threadIdx.x & 31;
    const int n0   = (threadIdx.x >> 5) << 4;      // wave's N-tile base
    const int m    = lane & 15;
    const int kb   = (lane >> 4) << 1;             // 0 (lanes 0-15) or 2 (lanes 16-31)

    v8f acc = {};
    #pragma unroll 8
    for (int k = 0; k < IN_CH; k += 4) {
        v2f a, b;
        // A 16x4: lanes 0-15 hold K=k+0,k+1 ; lanes 16-31 hold K=k+2,k+3
        a.x = xs[m * LDS_STRIDE + k + kb];
        a.y = xs[m * LDS_STRIDE + k + kb + 1];
        // B 4x16: row striped across lanes within a VGPR (mirrors A's K split)
        const float* Wp = W + (size_t)(k + kb) * HC + n0 + m;
        b.x = Wp[0];
        b.y = Wp[HC];
        acc = __builtin_amdgcn_wmma_f32_16x16x4_f32(false, a, false, b,
                                                    (short)0, acc, false, false);
    }

    // C/D layout: VGPR i -> M=i (lanes 0-15) / M=i+8 (lanes 16-31), N=lane&15
    const int mOff = (lane >> 4) << 3;
    const int col  = n0 + m;
    #pragma unroll
    for (int i = 0; i < 8; ++i) {
        int r = row0 + mOff + i;
        if (r < n) h[(size_t)r * HC + col] = acc[i];
    }
}

// ---------- 2: per-(node,head) attention logits ----------
__global__ void gat_scores(const float* __restrict__ h, const float* __restrict__ attn,
                           float* __restrict__ s_src, float* __restrict__ s_dst, int n) {
    int t = blockIdx.x * blockDim.x + threadIdx.x;
    if (t >= n * HEADS) return;
    int node = t >> 2, hd = t & 3;
    const float* hr = h + (size_t)node * HC + hd * OUT_CH;
    const float* as = attn + hd * (2 * OUT_CH);
    const float* ad = as + OUT_CH;
    float ss = 0.0f, sd = 0.0f;
    #pragma unroll 8
    for (int c = 0; c < OUT_CH; ++c) { float v = hr[c]; ss += v * as[c]; sd += v * ad[c]; }
    s_src[t] = ss; s_dst[t] = sd;
}

// ---------- 3: segment max over destination (row) ----------
__global__ void gat_segmax(const int* __restrict__ rows, const int* __restrict__ cols,
                           const float* __restrict__ s_src, const float* __restrict__ s_dst,
                           unsigned* __restrict__ segmax, int E) {
    int t = blockIdx.x * blockDim.x + threadIdx.x;
    if (t >= E * HEADS) return;
    int e = t >> 2, hd = t & 3;
    int r = rows[e], c = cols[e];
    float a = lrelu(s_src[r * HEADS + hd] + s_dst[c * HEADS + hd]);
    atomicMax(&segmax[r * HEADS + hd], mapf(a));
}

// ---------- 4: segment sum of exp(alpha - max) ----------
__global__ void gat_denom(const int* __restrict__ rows, const int* __restrict__ cols,
                          const float* __restrict__ s_src, const float* __restrict__ s_dst,
                          const unsigned* __restrict__ segmax, float* __restrict__ denom, int E) {
    int t = blockIdx.x * blockDim.x + threadIdx.x;
    if (t >= E * HEADS) return;
    int e = t >> 2, hd = t & 3;
    int r = rows[e], c = cols[e];
    float a  = lrelu(s_src[r * HEADS + hd] + s_dst[c * HEADS + hd]);
    float mx = unmapf(segmax[r * HEADS + hd]);
    atomicAdd(&denom[r * HEADS + hd], __expf(a - mx));
}

// ---------- 5: normalize + scatter-aggregate; one wave per edge ----------
__global__ void gat_scatter(const int* __restrict__ rows, const int* __restrict__ cols,
                            const float* __restrict__ s_src, const float* __restrict__ s_dst,
                            const unsigned* __restrict__ segmax, const float* __restrict__ denom,
                            const float* __restrict__ h,
                            float* __restrict__ out_mean, float* __restrict__ alpha_out, int E) {
    int e = blockIdx.x * (blockDim.x >> 5) + (threadIdx.x >> 5);
    if (e >= E) return;
    int lane = threadIdx.x & 31;
    int r = rows[e], c = cols[e];
    const float* hc = h + (size_t)c * HC;    // L2-resident (h = 25.6 MB < 192 MB L2)
    float val = 0.0f;
    #pragma unroll
    for (int hd = 0; hd < HEADS; ++hd) {
        float a  = lrelu(s_src[r * HEADS + hd] + s_dst[c * HEADS + hd]);
        float mx = unmapf(segmax[r * HEADS + hd]);
        float an = __expf(a - mx) / denom[r * HEADS + hd];
        if (lane == hd) alpha_out[(size_t)e * HEADS + hd] = an;
        val += an * hc[hd * OUT_CH + lane];
    }
    atomicAdd(&out_mean[(size_t)r * OUT_CH + lane], 0.25f * val);
}

extern "C" void kernel_launch(void* const* d_in, const int* in_sizes, int n_in,
                              void* d_out, int out_size, void* d_ws, size_t ws_size,
                              hipStream_t stream) {
    const float* x    = (const float*)d_in[0];
    const int*   ei   = (const int*)d_in[1];
    const float* W    = (const float*)d_in[2];
    const float* attn = (const float*)d_in[3];

    const int n = in_sizes[0] / IN_CH;
    const int E = in_sizes[1] / 2;
    const int* rows = ei;
    const int* cols = ei + E;

    // workspace layout (floats)
    float*    hbuf   = (float*)d_ws;
    float*    s_src  = hbuf  + (size_t)n * HC;
    float*    s_dst  = s_src + (size_t)n * HEADS;
    unsigned* segmax = (unsigned*)(s_dst + (size_t)n * HEADS);
    float*    denom  = (float*)(segmax + (size_t)n * HEADS);

    float* out_mean  = (float*)d_out;                 // [n,32]
    float* alpha_out = out_mean + (size_t)n * OUT_CH; // [E,4]

    gat_init<<<(n * OUT_CH + 255) / 256, 256, 0, stream>>>(out_mean, segmax, denom, n);

    gat_gemm<<<(n + 15) / 16, 256, 0, stream>>>(x, W, hbuf, n);

    gat_scores<<<(n * HEADS + 255) / 256, 256, 0, stream>>>(hbuf, attn, s_src, s_dst, n);

    gat_segmax<<<(E * HEADS + 255) / 256, 256, 0, stream>>>(rows, cols, s_src, s_dst, segmax, E);

    gat_denom<<<(E * HEADS + 255) / 256, 256, 0, stream>>>(rows, cols, s_src, s_dst, segmax, denom, E);

    gat_scatter<<<(E + 7) / 8, 256, 0, stream>>>(rows, cols, s_src, s_dst, segmax, denom,
                                                 hbuf, out_mean, alpha_out, E);
}